// NarrowAttentionLayer_11562051961157
// MI455X (gfx1250) — compile-verified
//
#include <hip/hip_runtime.h>

typedef __attribute__((ext_vector_type(2))) float v2f;
typedef __attribute__((ext_vector_type(8))) float v8f;

constexpr int B_   = 8;
constexpr int NA   = 4096;
constexpr int NB   = 1024;
constexpr int NK   = 128;
constexpr int NV   = 256;
constexpr int TOPK = 8;
constexpr int JT   = 64;   // query columns per workgroup
constexpr int AT   = 128;  // key rows per tile iteration
constexpr int SLD  = 68;   // score LDS row stride (padding kills bank conflicts)
constexpr float EPSF   = 1e-8f;
constexpr float NEGINF = -3.4028235e38f;

__global__ __launch_bounds__(256)
void narrow_attn_kernel(const float* __restrict__ key,
                        const float* __restrict__ query,
                        const float* __restrict__ value,
                        float* __restrict__ out)
{
    __shared__ float s_scores[AT * SLD];        // 34816 B
    __shared__ float s_mval[256 * TOPK];        // 8 KB
    __shared__ int   s_midx[256 * TOPK];        // 8 KB
    __shared__ float s_wv[JT * TOPK];           // 2 KB
    __shared__ int   s_wi[JT * TOPK];           // 2 KB

    const int tid   = threadIdx.x;
    const int wave  = tid >> 5;
    const int lane  = tid & 31;
    const int b     = blockIdx.x / (NB / JT);
    const int jBase = (blockIdx.x % (NB / JT)) * JT;

    const float scale = 0.0883883476483184f;    // 1/sqrt(128)

    const float* keyB = key   + (size_t)b * NA * NK;
    const float* qB   = query + (size_t)b * NB * NK;
    const float* valB = value + (size_t)b * NA * NV;

    float* out1  = out;                                              // (B,NV,NB)
    float* out_w = out + (size_t)B_ * NV * NB;                       // (B,TOPK,NB)
    int*   out_i = (int*)(out_w + (size_t)B_ * TOPK * NB);           // (B,TOPK,NB)

    // per-thread running top-8, ascending (tv[0] = current min)
    float tv[TOPK]; int ti[TOPK];
    #pragma unroll
    for (int i = 0; i < TOPK; ++i) { tv[i] = NEGINF; ti[i] = 0; }

    const int jScan   = tid & (JT - 1);
    const int quarter = tid >> 6;         // 0..3, each scans 32 of 128 tile rows

    // A-frag: lanes 0-15 rows M=0..15 K={0,1}; lanes 16-31 same rows K={2,3}
    const int fRow = lane & 15;
    const int kSub = (lane >> 4) * 2;

    for (int a0 = 0; a0 < NA; a0 += AT) {
        // ---------------- WMMA phase: wave computes 16(a) x 64(j) ----------
        const float* aPtr  = keyB + (size_t)(a0 + wave * 16 + fRow) * NK + kSub;
        const float* bPtr  = qB   + (size_t)(jBase + fRow) * NK + kSub;

        if (a0 + AT < NA)  // speculative prefetch of next key a-tile
            __builtin_prefetch(aPtr + (size_t)AT * NK, 0, 0);

        v8f acc0 = {}, acc1 = {}, acc2 = {}, acc3 = {};
        #pragma unroll 8
        for (int kk = 0; kk < NK; kk += 4) {
            v2f af = *(const v2f*)(aPtr + kk);
            v2f b0 = *(const v2f*)(bPtr + kk);
            v2f b1 = *(const v2f*)(bPtr + 16 * NK + kk);
            v2f b2 = *(const v2f*)(bPtr + 32 * NK + kk);
            v2f b3 = *(const v2f*)(bPtr + 48 * NK + kk);
            acc0 = __builtin_amdgcn_wmma_f32_16x16x4_f32(false, af, false, b0, (short)0, acc0, false, false);
            acc1 = __builtin_amdgcn_wmma_f32_16x16x4_f32(false, af, false, b1, (short)0, acc1, false, false);
            acc2 = __builtin_amdgcn_wmma_f32_16x16x4_f32(false, af, false, b2, (short)0, acc2, false, false);
            acc3 = __builtin_amdgcn_wmma_f32_16x16x4_f32(false, af, false, b3, (short)0, acc3, false, false);
        }

        // C layout: lanes 0-15 N=lane M=vv; lanes 16-31 N=lane-16 M=vv+8
        {
            const int m0  = (lane >> 4) * 8;
            const int col = lane & 15;
            #pragma unroll
            for (int vv = 0; vv < 8; ++vv) {
                const int row = wave * 16 + m0 + vv;
                s_scores[row * SLD +  0 + col] = acc0[vv] * scale;
                s_scores[row * SLD + 16 + col] = acc1[vv] * scale;
                s_scores[row * SLD + 32 + col] = acc2[vv] * scale;
                s_scores[row * SLD + 48 + col] = acc3[vv] * scale;
            }
        }
        __syncthreads();

        // ---------------- top-8 partial scan (4 threads per column) --------
        for (int r = 0; r < 32; ++r) {
            const int al = quarter * 32 + r;
            const float v = s_scores[al * SLD + jScan];
            if (v > tv[0]) {
                const int gi = a0 + al;
                int p = 0;
                #pragma unroll
                for (int i = 0; i < TOPK - 1; ++i) {
                    if (v > tv[i + 1]) { tv[i] = tv[i + 1]; ti[i] = ti[i + 1]; p = i + 1; }
                }
                tv[p] = v; ti[p] = gi;
            }
        }
        __syncthreads();
    }

    // ---------------- merge 4 partial top-8 per column, softmax ------------
    #pragma unroll
    for (int i = 0; i < TOPK; ++i) {
        s_mval[tid * TOPK + i] = tv[i];
        s_midx[tid * TOPK + i] = ti[i];
    }
    __syncthreads();

    if (tid < JT) {
        const int j = tid;
        float wsel[TOPK]; int isel[TOPK];
        for (int r = 0; r < TOPK; ++r) {
            float best = NEGINF; int bi = 0x7fffffff; int bslot = j * TOPK;
            for (int c = 0; c < 32; ++c) {
                const int slot = (j + JT * (c >> 3)) * TOPK + (c & 7);
                const float v = s_mval[slot];
                const int  id = s_midx[slot];
                if (v > best || (v == best && id < bi)) { best = v; bi = id; bslot = slot; }
            }
            wsel[r] = best; isel[r] = bi;
            s_mval[bslot] = NEGINF;                 // remove winner
        }
        // standardized softmax over the 8 (ddof=1, sigma/(1+sigma))
        float mean = 0.f;
        #pragma unroll
        for (int i = 0; i < TOPK; ++i) mean += wsel[i];
        mean *= (1.0f / TOPK);
        float var = 0.f;
        #pragma unroll
        for (int i = 0; i < TOPK; ++i) { const float d = wsel[i] - mean; var += d * d; }
        var *= (1.0f / (TOPK - 1));
        float sd = sqrtf(var) + EPSF;
        sd = sd / (1.0f + sd);
        const float inv = 1.0f / sd;
        const float zmax = (wsel[0] - mean) * inv;  // wsel sorted descending
        float e[TOPK]; float esum = 0.f;
        #pragma unroll
        for (int i = 0; i < TOPK; ++i) { e[i] = expf((wsel[i] - mean) * inv - zmax); esum += e[i]; }
        const float rs = 1.0f / esum;
        #pragma unroll
        for (int i = 0; i < TOPK; ++i) {
            const float wv = e[i] * rs;
            s_wv[j * TOPK + i] = wv;
            s_wi[j * TOPK + i] = isel[i];
            out_w[(size_t)(b * TOPK + i) * NB + jBase + j] = wv;
            out_i[(size_t)(b * TOPK + i) * NB + jBase + j] = isel[i];
        }
    }
    __syncthreads();

    // ---------------- value gather + layernorm (1 wave per column) --------
    for (int jg = 0; jg < JT; jg += 8) {
        const int j = jg + wave;
        float wv[TOPK]; int wi[TOPK];
        #pragma unroll
        for (int i = 0; i < TOPK; ++i) { wv[i] = s_wv[j * TOPK + i]; wi[i] = s_wi[j * TOPK + i]; }

        float vals[8];
        #pragma unroll
        for (int u = 0; u < 8; ++u) vals[u] = 0.f;
        #pragma unroll
        for (int i = 0; i < TOPK; ++i) {
            const float* vp = valB + (size_t)wi[i] * NV + lane;
            const float  wc = wv[i];
            #pragma unroll
            for (int u = 0; u < 8; ++u) vals[u] += wc * vp[u * 32];
        }
        float s = 0.f, s2 = 0.f;
        #pragma unroll
        for (int u = 0; u < 8; ++u) { s += vals[u]; s2 += vals[u] * vals[u]; }
        #pragma unroll
        for (int off = 16; off >= 1; off >>= 1) {
            s  += __shfl_xor(s,  off, 32);
            s2 += __shfl_xor(s2, off, 32);
        }
        const float mu  = s * (1.0f / NV);
        float var = (s2 - (float)NV * mu * mu) * (1.0f / (NV - 1));
        var = fmaxf(var, 0.f);
        const float inv = 1.0f / (sqrtf(var) + EPSF);
        #pragma unroll
        for (int u = 0; u < 8; ++u) {
            const int v = lane + 32 * u;
            out1[(size_t)b * NV * NB + (size_t)v * NB + jBase + j] = (vals[u] - mu) * inv;
        }
    }
}

extern "C" void kernel_launch(void* const* d_in, const int* in_sizes, int n_in,
                              void* d_out, int out_size, void* d_ws, size_t ws_size,
                              hipStream_t stream) {
    (void)in_sizes; (void)n_in; (void)out_size; (void)d_ws; (void)ws_size;
    const float* key   = (const float*)d_in[0];
    const float* query = (const float*)d_in[1];
    const float* value = (const float*)d_in[2];
    float* out = (float*)d_out;
    const int numWG = B_ * (NB / JT);   // 8 * 16 = 128 workgroups, 8 waves each
    narrow_attn_kernel<<<numWG, 256, 0, stream>>>(key, query, value, out);
}